// LQR_event_31868657336361
// MI455X (gfx1250) — compile-verified
//
#include <hip/hip_runtime.h>

#define DIM 100
#define LSTRIDE 114            // even stride: 8B-aligned b64 LDS frag loads, conflict-free banks
#define NT 7                   // ceil(100/16) output column tiles
#define WAVES 4
#define TPB (WAVES * 32)
#define TILES_PER_WAVE 4
#define ROWS_PER_BLOCK (WAVES * TILES_PER_WAVE * 16)   // 256

typedef __attribute__((ext_vector_type(2))) float v2f;
typedef __attribute__((ext_vector_type(8))) float v8f;

__global__ __launch_bounds__(TPB) void lqr_event_kernel(
    const float* __restrict__ state,   // (B, 200)
    const float* __restrict__ A,       // (100, 100)
    const float* __restrict__ K,       // (100, 100)
    const float* __restrict__ target,  // (100,)
    float* __restrict__ out,           // (B, 200)
    int nrows)
{
    __shared__ float sA[DIM * LSTRIDE];          // A row-major, padded stride
    __shared__ float sK[DIM * LSTRIDE];
    __shared__ float sTg[DIM];
    __shared__ float sSat[WAVES][16 * LSTRIDE];  // per-wave staging: sat(x)
    __shared__ float sT[WAVES][16 * LSTRIDE];    // per-wave staging: x + e_x - target

    const int tid  = threadIdx.x;
    const int wave = tid >> 5;
    const int lane = tid & 31;
    const int lhalf = lane >> 4;   // 0: lanes 0-15, 1: lanes 16-31
    const int l16   = lane & 15;

    // ---- stage A, K, target into LDS (hits L2 on every block; 80KB resident) ----
    for (int idx = tid; idx < DIM * DIM; idx += TPB) {
        int r = idx / DIM, c = idx - r * DIM;
        sA[r * LSTRIDE + c] = A[idx];
        sK[r * LSTRIDE + c] = K[idx];
    }
    for (int c = tid; c < DIM; c += TPB) sTg[c] = target[c];
    __syncthreads();

    const long blockRow0 = (long)blockIdx.x * ROWS_PER_BLOCK;

    for (int t = 0; t < TILES_PER_WAVE; ++t) {
        const long rowBase = blockRow0 + (long)(wave * TILES_PER_WAVE + t) * 16;
        if (rowBase + 16 > (long)nrows) continue;   // wave-uniform guard

        // ---- stage sat and t for this 16-row tile (coalesced global reads) ----
        for (int r = 0; r < 16; ++r) {
            const long g = (rowBase + r) * (2 * DIM);
            const int lrow = r * LSTRIDE;
            for (int c = lane; c < DIM; c += 32) {
                float x  = state[g + c];
                float ex = state[g + DIM + c];
                float x2 = x * x;
                float s  = x2 / (1.0f + x2);
                sSat[wave][lrow + c] = s;
                sT[wave][lrow + c]   = (x + ex) - sTg[c];
            }
        }
        // per-wave LDS: DS ops from the same wave are in-order; compiler inserts s_wait_dscnt

        // ---- prefetch next row-tile (contiguous 12.8KB) while GEMM runs ----
        const long nextBase = rowBase + 16;
        if (t + 1 < TILES_PER_WAVE && nextBase + 16 <= (long)nrows) {
            const float* pf = state + nextBase * (2 * DIM);
            for (int i = lane; i < 100; i += 32)          // 100 x 128B lines
                __builtin_prefetch(pf + (long)i * 32, 0, 1); // global_prefetch_b8
        }

        // ---- GEMM over 7 output column tiles ----
        for (int nt = 0; nt < NT; ++nt) {
            const int n  = nt * 16 + l16;
            const int nr = (n < DIM) ? n : (DIM - 1);   // clamped rows feed dropped outputs

            // A-frag: lane holds sat[row=l16][k0+2*lhalf .. +1]  (16x4 layout)
            const float* pSat = &sSat[wave][l16 * LSTRIDE + 2 * lhalf];
            const float* pT   = &sT[wave][l16 * LSTRIDE + 2 * lhalf];
            // B-frag: lane holds Aᵀ[k0+2*lhalf .. +1][n] = A[n][k0+2*lhalf .. +1] (4x16 layout)
            const float* pA   = &sA[nr * LSTRIDE + 2 * lhalf];
            const float* pK   = &sK[nr * LSTRIDE + 2 * lhalf];

            v8f acc1 = {0.f, 0.f, 0.f, 0.f, 0.f, 0.f, 0.f, 0.f};  // sat @ Aᵀ
            v8f acc2 = {0.f, 0.f, 0.f, 0.f, 0.f, 0.f, 0.f, 0.f};  // t   @ Kᵀ

            // unroll 5: enough DS-load lead distance for the serialized WMMA
            // accumulator chains, while keeping the wave under 256 VGPRs
            // (no s_set_vgpr_msb traffic in the hot loop).
#pragma unroll 5
            for (int k0 = 0; k0 < DIM; k0 += 4) {   // 25 steps cover k=0..99
                v2f aS = *(const v2f*)(pSat + k0);  // ds_load_b64 (8B aligned)
                v2f aT = *(const v2f*)(pT + k0);
                v2f bA = *(const v2f*)(pA + k0);
                v2f bK = *(const v2f*)(pK + k0);
                acc1 = __builtin_amdgcn_wmma_f32_16x16x4_f32(
                    false, aS, false, bA, (short)0, acc1, false, false);
                acc2 = __builtin_amdgcn_wmma_f32_16x16x4_f32(
                    false, aT, false, bK, (short)0, acc2, false, false);
            }

            // ---- fused epilogue: dx = -x + acc1 - acc2*sat ; store [dx, -dx] ----
            const int e = nt * 16 + l16;
            if (e < DIM) {
#pragma unroll
                for (int v = 0; v < 8; ++v) {
                    const int lr = v + 8 * lhalf;          // C/D layout: M = v + 8*(lane>=16)
                    const long row = rowBase + lr;
                    float x  = state[row * (2 * DIM) + e]; // L2/WGP$ hit
                    float sv = sSat[wave][lr * LSTRIDE + e];
                    float dx = acc1[v] - acc2[v] * sv - x;
                    out[row * (2 * DIM) + e]       = dx;
                    out[row * (2 * DIM) + DIM + e] = -dx;
                }
            }
        }
    }
}

extern "C" void kernel_launch(void* const* d_in, const int* in_sizes, int n_in,
                              void* d_out, int out_size, void* d_ws, size_t ws_size,
                              hipStream_t stream) {
    const float* state  = (const float*)d_in[0];
    const float* A      = (const float*)d_in[1];
    const float* K      = (const float*)d_in[2];
    const float* target = (const float*)d_in[3];
    float* out = (float*)d_out;

    const int nrows  = in_sizes[0] / (2 * DIM);                 // 262144
    const int blocks = (nrows + ROWS_PER_BLOCK - 1) / ROWS_PER_BLOCK;  // 1024

    lqr_event_kernel<<<blocks, TPB, 0, stream>>>(state, A, K, target, out, nrows);
}